// Network_44865228374080
// MI455X (gfx1250) — compile-verified
//
#include <hip/hip_runtime.h>
#include <hip/hip_bf16.h>

// ---------------------------------------------------------------------------
// Problem constants (from the reference): B=256, T=4096, N_OBS=64, N_ACT=18, H=8
// scan axis is B (256 sequential steps), 4096 independent recurrent lanes.
// ---------------------------------------------------------------------------
#define BDIM 256
#define TDIM 4096
#define NOBS 64
#define NACT 18
#define HDIM 8

typedef __attribute__((ext_vector_type(2))) float v2f;
typedef __attribute__((ext_vector_type(8))) float v8f;

// D = A(16x4 f32) * B(4x16 f32) + C(16x16 f32)
// If this builtin is missing on the device side, clang's deferred diagnostics
// will report it when compiling for gfx1250 (the host pass ignores it).
__device__ __forceinline__ v8f wmma4(v2f a, v2f b, v8f c) {
  return __builtin_amdgcn_wmma_f32_16x16x4_f32(
      /*neg_a=*/false, a, /*neg_b=*/false, b,
      /*c_mod=*/(short)0, c, /*reuse_a=*/false, /*reuse_b=*/false);
}

__device__ __forceinline__ float sigm_(float x) {
  float e = __expf(-x);
  return 1.0f / (1.0f + e);
}
__device__ __forceinline__ float tanh_(float x) {
  float e = __expf(-2.0f * x);
  return (1.0f - e) / (1.0f + e);
}

// ---------------------------------------------------------------------------
// Kernel 1: x2[row][0:8] = relu(relu(states[row]·W1^T + b1)·W2^T + b2)
// One wave per 16-row tile. Layer1: 16x wmma (K=64). Layer2: 1x wmma (K=4).
// Weights live in registers in WMMA B-layout; x1 C->A transpose goes via LDS.
// ---------------------------------------------------------------------------
__global__ __launch_bounds__(256) void frontend_kernel(
    const float* __restrict__ states, const float* __restrict__ W1,
    const float* __restrict__ b1, const float* __restrict__ W2,
    const float* __restrict__ b2, float* __restrict__ x2) {
  __shared__ float Xs[8 * 64];  // per-wave 16x4 transpose slice
  const int tid = threadIdx.x;
  const int w = tid >> 5, l = tid & 31;
  const int n = l & 15;   // N column / M row within halves
  const int hi = l >> 4;  // lane half (K offset +2 for A/B layouts)
  float* X = Xs + w * 64;

  // W1 (4x64) in B-layout per K-block: .x -> k=kb*4+2*hi, .y -> +1 ; cols n<4
  v2f w1b[16];
#pragma unroll
  for (int kb = 0; kb < 16; ++kb) {
    const int k = kb * 4 + 2 * hi;
    w1b[kb].x = (n < 4) ? W1[n * NOBS + k] : 0.0f;
    w1b[kb].y = (n < 4) ? W1[n * NOBS + k + 1] : 0.0f;
  }
  // W2 (8x4) in B-layout (single K-block), cols n<8
  v2f w2b;
  w2b.x = (n < 8) ? W2[n * 4 + 2 * hi] : 0.0f;
  w2b.y = (n < 8) ? W2[n * 4 + 2 * hi + 1] : 0.0f;
  const float bias1 = (n < 4) ? b1[n] : 0.0f;
  const float bias2 = (n < 8) ? b2[n] : 0.0f;

  const long tile = (long)blockIdx.x * 8 + w;
  const long row0 = tile * 16;
  // A-layout: lane holds row m=n, K pair (2*hi, 2*hi+1) within each K-block
  const float* srow = states + (row0 + n) * (long)NOBS + 2 * hi;

  v8f acc;
#pragma unroll
  for (int r = 0; r < 8; ++r) acc[r] = bias1;
#pragma unroll
  for (int kb = 0; kb < 16; ++kb) {
    v2f a = *(const v2f*)(srow + kb * 4);
    acc = wmma4(a, w1b[kb], acc);
  }
#pragma unroll
  for (int r = 0; r < 8; ++r) acc[r] = fmaxf(acc[r], 0.0f);  // relu1

  // C-layout -> A-layout transpose of x1 (16 rows x 4) through LDS
  if (n < 4) {
#pragma unroll
    for (int r = 0; r < 8; ++r) X[(r + 8 * hi) * 4 + n] = acc[r];
  }
  __builtin_amdgcn_wave_barrier();
  v2f a2 = *(const v2f*)(X + n * 4 + 2 * hi);

  v8f acc2;
#pragma unroll
  for (int r = 0; r < 8; ++r) acc2[r] = bias2;
  acc2 = wmma4(a2, w2b, acc2);
#pragma unroll
  for (int r = 0; r < 8; ++r) acc2[r] = fmaxf(acc2[r], 0.0f);  // relu2

  // store x2 (row-major 1M x 8), valid cols n<8
  if (n < 8) {
#pragma unroll
    for (int r = 0; r < 8; ++r)
      x2[(row0 + r + 8 * hi) * (long)HDIM + n] = acc2[r];
  }
}

// ---------------------------------------------------------------------------
// Kernel 2: 256-step LSTM recurrence over batch axis + action/value heads.
// One wave per 16 t-lanes (256 waves total). Per step:
//   gates(16x32) = x·Wih^T + h·Whh^T + biases  -> 8 wmmas (2 N-halves, K=8+8)
//   activations + state update in C-layout (shfl_xor(8) pairs i/f and g/o)
//   heads(16x19) = h·[Wa;Wv]^T -> 4 wmmas, stored straight to d_out.
// ---------------------------------------------------------------------------
__global__ __launch_bounds__(256) void lstm_kernel(
    const float* __restrict__ x2, const unsigned char* __restrict__ dones,
    const float* __restrict__ hx, const float* __restrict__ Wih,
    const float* __restrict__ bih, const float* __restrict__ Whh,
    const float* __restrict__ bhh, const float* __restrict__ Wa,
    const float* __restrict__ ba, const float* __restrict__ Wv,
    const float* __restrict__ bv, float* __restrict__ act,
    float* __restrict__ val) {
  __shared__ float Hss[8 * 160];  // per-wave 16 rows x stride-10 h staging
  const int tid = threadIdx.x;
  const int w = tid >> 5, l = tid & 31;
  const int n = l & 15;
  const int hi = l >> 4;
  float* Hs = Hss + w * 160;
  const int t0 = (blockIdx.x * 8 + w) * 16;

  // Gate weights in B-layout: [ih/hh][N-half p][K-block q]
  v2f wb[2][2][2];
  float gb[2];
#pragma unroll
  for (int p = 0; p < 2; ++p) {
    const int gate = p * 16 + n;  // cols: p0 -> i,f ; p1 -> g,o
#pragma unroll
    for (int q = 0; q < 2; ++q) {
      const int k = q * 4 + 2 * hi;
      wb[0][p][q].x = Wih[gate * HDIM + k];
      wb[0][p][q].y = Wih[gate * HDIM + k + 1];
      wb[1][p][q].x = Whh[gate * HDIM + k];
      wb[1][p][q].y = Whh[gate * HDIM + k + 1];
    }
    gb[p] = bih[gate] + bhh[gate];
  }
  // Head weights: group0 = Wa rows 0..15; group1 cols: 0,1 = Wa rows 16,17; 2 = Wv
  v2f waB[2][2];
  float hb[2];
#pragma unroll
  for (int q = 0; q < 2; ++q) {
    const int k = q * 4 + 2 * hi;
    waB[0][q].x = Wa[n * HDIM + k];
    waB[0][q].y = Wa[n * HDIM + k + 1];
    float bx = 0.0f, by = 0.0f;
    if (n < 2) {
      bx = Wa[(16 + n) * HDIM + k];
      by = Wa[(16 + n) * HDIM + k + 1];
    } else if (n == 2) {
      bx = Wv[k];
      by = Wv[k + 1];
    }
    waB[1][q].x = bx;
    waB[1][q].y = by;
  }
  hb[0] = ba[n];
  hb[1] = (n < 2) ? ba[16 + n] : ((n == 2) ? bv[0] : 0.0f);

  // Recurrent state in C-layout: element (row m=r+8*hi, hidden j=n) valid n<8
  float h[8], c[8];
  const int j = n & 7;  // keep garbage lanes in-bounds
#pragma unroll
  for (int r = 0; r < 8; ++r) {
    const int m = r + 8 * hi;
    h[r] = hx[(size_t)(t0 + m) * HDIM + j];
    c[r] = hx[(size_t)TDIM * HDIM + (size_t)(t0 + m) * HDIM + j];
  }

  for (int b = 0; b < BDIM; ++b) {
    const size_t rb = (size_t)b * TDIM + t0;

    // x2 input in A-layout (row m=n, K pairs)
    const float* xrow = x2 + (rb + n) * HDIM + 2 * hi;
    v2f ax0 = *(const v2f*)(xrow);
    v2f ax1 = *(const v2f*)(xrow + 4);

    // done-mask per row (bool bytes); reset h,c before the gates
    const unsigned char* dp = dones + rb + 8 * hi;
    const unsigned int d0 = *(const unsigned int*)(dp);
    const unsigned int d1 = *(const unsigned int*)(dp + 4);
#pragma unroll
    for (int r = 0; r < 8; ++r) {
      const unsigned int byt = ((r < 4 ? d0 : d1) >> ((r & 3) * 8)) & 0xffu;
      const float keep = byt ? 0.0f : 1.0f;
      h[r] *= keep;
      c[r] *= keep;
    }

    // stage masked h: C-layout -> A-layout via LDS
    if (n < 8) {
#pragma unroll
      for (int r = 0; r < 8; ++r) Hs[(r + 8 * hi) * 10 + n] = h[r];
    }
    __builtin_amdgcn_wave_barrier();
    v2f ah0 = *(const v2f*)(Hs + n * 10 + 2 * hi);
    v2f ah1 = *(const v2f*)(Hs + n * 10 + 4 + 2 * hi);

    v8f g0, g1;
#pragma unroll
    for (int r = 0; r < 8; ++r) {
      g0[r] = gb[0];
      g1[r] = gb[1];
    }
    g0 = wmma4(ax0, wb[0][0][0], g0);
    g0 = wmma4(ax1, wb[0][0][1], g0);
    g0 = wmma4(ah0, wb[1][0][0], g0);
    g0 = wmma4(ah1, wb[1][0][1], g0);
    g1 = wmma4(ax0, wb[0][1][0], g1);
    g1 = wmma4(ax1, wb[0][1][1], g1);
    g1 = wmma4(ah0, wb[1][1][0], g1);
    g1 = wmma4(ah1, wb[1][1][1], g1);

    // activations + state update: cols<8 hold i(g0)/g(g1); cols>=8 hold f/o
#pragma unroll
    for (int r = 0; r < 8; ++r) {
      const float s0 = sigm_(g0[r]);                              // i or f
      const float a1 = (n < 8) ? tanh_(g1[r]) : sigm_(g1[r]);     // g or o
      const float f_ = __shfl_xor(s0, 8, 32);
      const float o_ = __shfl_xor(a1, 8, 32);
      const float cn = f_ * c[r] + s0 * a1;
      c[r] = cn;
      h[r] = o_ * tanh_(cn);
    }

    // stage new h for the head GEMM (and reused next step after masking)
    __builtin_amdgcn_wave_barrier();
    if (n < 8) {
#pragma unroll
      for (int r = 0; r < 8; ++r) Hs[(r + 8 * hi) * 10 + n] = h[r];
    }
    __builtin_amdgcn_wave_barrier();
    v2f ha0 = *(const v2f*)(Hs + n * 10 + 2 * hi);
    v2f ha1 = *(const v2f*)(Hs + n * 10 + 4 + 2 * hi);

    v8f A0, A1;
#pragma unroll
    for (int r = 0; r < 8; ++r) {
      A0[r] = hb[0];
      A1[r] = hb[1];
    }
    A0 = wmma4(ha0, waB[0][0], A0);
    A0 = wmma4(ha1, waB[0][1], A0);
    A1 = wmma4(ha0, waB[1][0], A1);
    A1 = wmma4(ha1, waB[1][1], A1);

    // actions cols 0..15 : every lane valid
#pragma unroll
    for (int r = 0; r < 8; ++r) {
      const size_t row = rb + r + 8 * hi;
      act[row * NACT + n] = A0[r];
    }
    // actions cols 16,17 + value (col 2 of group1)
#pragma unroll
    for (int r = 0; r < 8; ++r) {
      const size_t row = rb + r + 8 * hi;
      if (n < 2) {
        act[row * NACT + 16 + n] = A1[r];
      } else if (n == 2) {
        val[row] = A1[r];
      }
    }
  }
}

// ---------------------------------------------------------------------------
extern "C" void kernel_launch(void* const* d_in, const int* in_sizes, int n_in,
                              void* d_out, int out_size, void* d_ws,
                              size_t ws_size, hipStream_t stream) {
  (void)in_sizes;
  (void)n_in;
  (void)out_size;
  (void)ws_size;
  const float* states = (const float*)d_in[0];
  const unsigned char* dones = (const unsigned char*)d_in[1];  // bool bytes
  const float* hx = (const float*)d_in[2];
  const float* W1 = (const float*)d_in[3];
  const float* b1 = (const float*)d_in[4];
  const float* W2 = (const float*)d_in[5];
  const float* b2 = (const float*)d_in[6];
  const float* Wih = (const float*)d_in[7];
  const float* bih = (const float*)d_in[8];
  const float* Whh = (const float*)d_in[9];
  const float* bhh = (const float*)d_in[10];
  const float* Wa = (const float*)d_in[11];
  const float* ba = (const float*)d_in[12];
  const float* Wv = (const float*)d_in[13];
  const float* bv = (const float*)d_in[14];

  float* act = (float*)d_out;                                // (B*T, 18) flat
  float* val = act + (size_t)BDIM * TDIM * NACT;             // (B*T,) flat
  float* x2 = (float*)d_ws;  // needs B*T*8*4 = 32 MB of scratch

  // Phase 1: MLP frontend, 65536 16-row tiles, 8 waves per block.
  frontend_kernel<<<8192, 256, 0, stream>>>(states, W1, b1, W2, b2, x2);
  // Phase 2: 256 waves (16 t-lanes each) run the 256-step recurrence + heads.
  lstm_kernel<<<32, 256, 0, stream>>>(x2, dones, hx, Wih, bih, Whh, bhh, Wa,
                                      ba, Wv, bv, act, val);
}